// MemNetMasked_47150150975832
// MI455X (gfx1250) — compile-verified
//
#include <hip/hip_runtime.h>
#include <hip/hip_bf16.h>

// ---------------------------------------------------------------------------
// Sizes (fixed by the reference)
// ---------------------------------------------------------------------------
#define BB   16      // batch
#define NN   127     // nodes
#define SS   32      // sentence
#define EE   16      // embedding
#define MM   128     // memory slots (== NN+1, so pad_m == 0)
#define LL   128     // linear output
#define VV   128     // vocab
#define BT   (BB * (NN + 1))   // 2048 rows

typedef __attribute__((ext_vector_type(2))) float v2f;
typedef __attribute__((ext_vector_type(8))) float v8f;

// ---------------------------------------------------------------------------
// Kernel 1: per-row memory network up to h = u + o   (no WMMA; gather-bound)
// One block per bt row, 128 threads (thread == memory slot m).
// ---------------------------------------------------------------------------
__global__ __launch_bounds__(128)
void memnet_row_kernel(const float* __restrict__ node_fts,   // [B,N,S]
                       const float* __restrict__ edge_fts,   // [B,N,N,S]
                       const float* __restrict__ graph_fts,  // [B,S]
                       const float* __restrict__ adj_mat,    // [B,N,N]
                       const float* __restrict__ enc,        // [S,E]
                       const float* __restrict__ query_b,    // [V-1,E]
                       const float* __restrict__ stories_b,  // [V-1,E]
                       const float* __restrict__ output_b,   // [V-1,E]
                       const float* __restrict__ mem_c,      // [M,E]
                       float* __restrict__ hbuf)             // [BT,E]
{
    __shared__ float sb_l[VV * EE];    // stories table (+nil row)
    __shared__ float ob_l[VV * EE];    // output table  (+nil row)
    __shared__ float qb_l[VV * EE];    // query table   (+nil row)
    __shared__ float enc_l[SS * EE];
    __shared__ float cj_l[MM * EE];    // Cj[m][e]
    __shared__ float u_l[EE];
    __shared__ int   qidx_l[SS];
    __shared__ float lg_l[MM];
    __shared__ float p_l[MM];
    __shared__ float red[MM];

    const int tid = threadIdx.x;       // 0..127
    const int bt  = blockIdx.x;        // 0..2047
    const int b   = bt >> 7;           // N+1 == 128
    const int n   = bt & 127;

    // ---- stage vocab tables + enc into LDS (nil row = 0) ----
    for (int i = tid; i < VV * EE; i += 128) {
        const bool real = (i < (VV - 1) * EE);
        sb_l[i] = real ? stories_b[i] : 0.f;
        ob_l[i] = real ? output_b[i]  : 0.f;
        qb_l[i] = real ? query_b[i]   : 0.f;
    }
    for (int i = tid; i < SS * EE; i += 128) enc_l[i] = enc[i];
    if (tid < SS) {
        float qv = (n < NN) ? node_fts[((size_t)(b * NN + n)) * SS + tid]
                            : graph_fts[b * SS + tid];
        int qi = (int)qv;
        qidx_l[tid] = min(max(qi, 0), VV - 1);
    }
    __syncthreads();

    // ---- u[e] = sum_s qb[qidx[s],e] * enc[s,e]  (threads 0..15) ----
    if (tid < EE) {
        float acc = 0.f;
        for (int s = 0; s < SS; ++s)
            acc += qb_l[qidx_l[s] * EE + tid] * enc_l[s * EE + tid];
        u_l[tid] = acc;
    }

    // ---- mem/Cj accumulation for slot m = tid ----
    const int m = tid;
    float macc[EE], cacc[EE];
#pragma unroll
    for (int e = 0; e < EE; ++e) { macc[e] = 0.f; cacc[e] = 0.f; }

    const bool pad = (n == NN) || (m == NN);
    float adjv = 0.f;
    const float4* ef4 = nullptr;
    if (!pad) {
        const size_t eoff = (((size_t)(b * NN + n)) * NN + m) * SS;
        ef4  = (const float4*)(edge_fts + eoff);            // 128B aligned
        adjv = adj_mat[(size_t)(b * NN + n) * NN + m];
    }
    const bool zero_story = pad || (adjv == 0.f);

    for (int s4 = 0; s4 < SS / 4; ++s4) {
        float vals[4] = {0.f, 0.f, 0.f, 0.f};
        if (!zero_story) {
            float4 v = ef4[s4];
            vals[0] = v.x; vals[1] = v.y; vals[2] = v.z; vals[3] = v.w;
        }
#pragma unroll
        for (int si = 0; si < 4; ++si) {
            const int s = s4 * 4 + si;
            int idx = zero_story ? 0 : min(max((int)vals[si], 0), VV - 1);
            const float4* sr = (const float4*)(sb_l + idx * EE);
            const float4* orr = (const float4*)(ob_l + idx * EE);
            const float4* er = (const float4*)(enc_l + s * EE);
#pragma unroll
            for (int q = 0; q < 4; ++q) {
                float4 sv = sr[q], ov = orr[q], ev = er[q];
                macc[4*q+0] += sv.x * ev.x;  cacc[4*q+0] += ov.x * ev.x;
                macc[4*q+1] += sv.y * ev.y;  cacc[4*q+1] += ov.y * ev.y;
                macc[4*q+2] += sv.z * ev.z;  cacc[4*q+2] += ov.z * ev.z;
                macc[4*q+3] += sv.w * ev.w;  cacc[4*q+3] += ov.w * ev.w;
            }
        }
    }

    // stash Cj, then (after barrier) logit = dot(mem + memory_contents, u)
#pragma unroll
    for (int e = 0; e < EE; ++e) cj_l[m * EE + e] = cacc[e];
    __syncthreads();   // u_l and cj_l visible

    float lg = 0.f;
#pragma unroll
    for (int e = 0; e < EE; ++e)
        lg += (macc[e] + mem_c[m * EE + e]) * u_l[e];
    lg_l[m] = lg;
    red[m]  = lg;
    __syncthreads();

    // ---- softmax over 128 slots: tree max, exp, tree sum ----
    for (int off = 64; off >= 1; off >>= 1) {
        if (tid < off) red[tid] = fmaxf(red[tid], red[tid + off]);
        __syncthreads();
    }
    const float mx = red[0];
    __syncthreads();
    const float ex = expf(lg_l[m] - mx);
    p_l[m] = ex;
    red[m] = ex;
    __syncthreads();
    for (int off = 64; off >= 1; off >>= 1) {
        if (tid < off) red[tid] += red[tid + off];
        __syncthreads();
    }
    const float rden = 1.f / red[0];

    // ---- o[e] = sum_m Cj[m,e] * p[m];  h = u + o ----
    if (tid < EE) {
        float o = 0.f;
        for (int mm = 0; mm < MM; ++mm)
            o += cj_l[mm * EE + tid] * (p_l[mm] * rden);
        hbuf[(size_t)bt * EE + tid] = u_l[tid] + o;
    }
}

// ---------------------------------------------------------------------------
// Kernel 2: dense layers via WMMA f32 16x16x4.
//   X   = relu(H[16x16] @ W_out[16x128])
//   RET = X[16x128] @ W_fin[128x128]
// One block per 16 bt rows; 4 waves, each wave owns two 16-wide N-tiles.
// Fragment layouts per CDNA5 ISA §7.12.2:
//   A f32 16x4 : lane l -> M = l&15 ; v0 = K(2g), v1 = K(2g+1), g = l>>4
//   B f32 4x16 : lane l -> N = l&15 ; v0 = K(2g), v1 = K(2g+1)
//   C/D 16x16  : lane l -> N = l&15 ; VGPR r -> M = r + 8*(l>>4)
// ---------------------------------------------------------------------------
__global__ __launch_bounds__(128)
void memnet_gemm_kernel(const float* __restrict__ hbuf,   // [BT,16]
                        const float* __restrict__ W_out,  // [16,128]
                        const float* __restrict__ W_fin,  // [128,128]
                        float* __restrict__ retbuf)       // [BT,128]
{
    __shared__ float x_l[16 * LL];     // X[m][j]

    const int tid  = threadIdx.x;
    const int wave = tid >> 5;
    const int lane = tid & 31;
    const int lm   = lane & 15;
    const int lg   = lane >> 4;        // half-wave selector
    const int r0   = blockIdx.x * 16;  // base bt row

    // ---- GEMM1 + ReLU: X = relu(H @ W_out), K = 16 (4 WMMA steps/tile) ----
    for (int t = 0; t < 2; ++t) {
        const int n0 = wave * 32 + t * 16;
        v8f c = {};
#pragma unroll
        for (int kk = 0; kk < EE; kk += 4) {
            v2f a, bf;
            a.x  = hbuf[(size_t)(r0 + lm) * EE + kk + 2 * lg];
            a.y  = hbuf[(size_t)(r0 + lm) * EE + kk + 2 * lg + 1];
            bf.x = W_out[(kk + 2 * lg)     * LL + n0 + lm];
            bf.y = W_out[(kk + 2 * lg + 1) * LL + n0 + lm];
            c = __builtin_amdgcn_wmma_f32_16x16x4_f32(
                    false, a, false, bf, (short)0, c, false, false);
        }
#pragma unroll
        for (int r = 0; r < 8; ++r) {
            const int mrow = r + 8 * lg;
            float v = c[r];
            x_l[mrow * LL + n0 + lm] = v > 0.f ? v : 0.f;
        }
    }
    __syncthreads();

    // ---- GEMM2: RET = X @ W_fin, K = 128 (32 WMMA steps/tile) ----
    for (int t = 0; t < 2; ++t) {
        const int n0 = wave * 32 + t * 16;
        v8f c = {};
#pragma unroll
        for (int kk = 0; kk < LL; kk += 4) {
            v2f a, bf;
            a.x  = x_l[lm * LL + kk + 2 * lg];
            a.y  = x_l[lm * LL + kk + 2 * lg + 1];
            bf.x = W_fin[(kk + 2 * lg)     * VV + n0 + lm];
            bf.y = W_fin[(kk + 2 * lg + 1) * VV + n0 + lm];
            c = __builtin_amdgcn_wmma_f32_16x16x4_f32(
                    false, a, false, bf, (short)0, c, false, false);
        }
#pragma unroll
        for (int r = 0; r < 8; ++r) {
            const int mrow = r + 8 * lg;
            retbuf[(size_t)(r0 + mrow) * VV + n0 + lm] = c[r];
        }
    }
}

// ---------------------------------------------------------------------------
// Kernel 3: out[b,n,v] = ret[b,n,v] + ret[b,127,v]
// ---------------------------------------------------------------------------
__global__ __launch_bounds__(256)
void memnet_combine_kernel(const float* __restrict__ retbuf,  // [BT,128]
                           float* __restrict__ out)           // [B,N,V]
{
    const int idx = blockIdx.x * 256 + threadIdx.x;
    if (idx >= BB * NN * VV) return;
    const int v = idx & (VV - 1);
    const int n = (idx >> 7) % NN;
    const int b = idx / (NN * VV);
    out[idx] = retbuf[(size_t)(b * (NN + 1) + n)  * VV + v] +
               retbuf[(size_t)(b * (NN + 1) + NN) * VV + v];
}

// ---------------------------------------------------------------------------
// Launch
// Input order: 0 node_fts, 1 edge_fts, 2 graph_fts, 3 adj_mat, 4 hidden,
//              5 enc, 6 query_biases, 7 stories_biases, 8 output_biases,
//              9 memory_contents, 10 W_int, 11 W_out, 12 W_fin
// ---------------------------------------------------------------------------
extern "C" void kernel_launch(void* const* d_in, const int* in_sizes, int n_in,
                              void* d_out, int out_size, void* d_ws, size_t ws_size,
                              hipStream_t stream) {
    const float* node_fts  = (const float*)d_in[0];
    const float* edge_fts  = (const float*)d_in[1];
    const float* graph_fts = (const float*)d_in[2];
    const float* adj_mat   = (const float*)d_in[3];
    const float* enc       = (const float*)d_in[5];
    const float* query_b   = (const float*)d_in[6];
    const float* stories_b = (const float*)d_in[7];
    const float* output_b  = (const float*)d_in[8];
    const float* mem_c     = (const float*)d_in[9];
    const float* W_out     = (const float*)d_in[11];
    const float* W_fin     = (const float*)d_in[12];
    float* out = (float*)d_out;

    float* hbuf   = (float*)d_ws;          // [BT,16]  = 128 KB
    float* retbuf = hbuf + (size_t)BT * EE; // [BT,128] = 1 MB

    memnet_row_kernel<<<BT, 128, 0, stream>>>(
        node_fts, edge_fts, graph_fts, adj_mat, enc,
        query_b, stories_b, output_b, mem_c, hbuf);

    memnet_gemm_kernel<<<BT / 16, 128, 0, stream>>>(hbuf, W_out, W_fin, retbuf);

    const int total = BB * NN * VV;
    memnet_combine_kernel<<<(total + 255) / 256, 256, 0, stream>>>(retbuf, out);
}